// ForwardKinematics_71966472011824
// MI455X (gfx1250) — compile-verified
//
#include <hip/hip_runtime.h>
#include <stdint.h>

#define NQ 7
#define BLOCK 256
// Per-joint constant block layout in d_ws (floats):
//  [0..8]  P  = Rj + Rj*W^2/n^2
//  [9..17] Q  = Rj*W/n
//  [18..26]Rm = -Rj*W^2/n^2
//  [27..29]t  = joint_xyz
//  [30]    n  = |axis|
//  [31]    pad
#define CST_STRIDE 32
#define CST_LINK   (NQ * CST_STRIDE)   // link translation (3 floats) after joints

typedef __attribute__((ext_vector_type(4))) unsigned int v4u;
typedef __attribute__((ext_vector_type(4))) int          v4i;
typedef __attribute__((ext_vector_type(8))) int          v8i;
typedef __attribute__((ext_vector_type(4))) float        v4f;

// ---------------- setup: fold per-joint constants (lane j = joint j) -------

__device__ inline void mm3(const float* A, const float* B, float* C) {
#pragma unroll
  for (int r = 0; r < 3; ++r)
#pragma unroll
    for (int c = 0; c < 3; ++c)
      C[3 * r + c] = A[3 * r + 0] * B[0 + c] +
                     A[3 * r + 1] * B[3 + c] +
                     A[3 * r + 2] * B[6 + c];
}

__global__ void fk_setup_kernel(const float* __restrict__ jxyz,
                                const float* __restrict__ jrpy,
                                const float* __restrict__ jaxis,
                                const float* __restrict__ lxyz,
                                const int* __restrict__ idx,
                                float* __restrict__ cst) {
  if (blockIdx.x != 0) return;
  const int j = (int)threadIdx.x;

  if (j < NQ) {
    const float r = jrpy[3 * j + 0], p = jrpy[3 * j + 1], y = jrpy[3 * j + 2];
    float sr, cr, sp, cp, sy, cy;
    sincosf(r, &sr, &cr);
    sincosf(p, &sp, &cp);
    sincosf(y, &sy, &cy);
    const float Rx[9] = {1, 0, 0, 0, cr, -sr, 0, sr, cr};
    const float Ry[9] = {cp, 0, sp, 0, 1, 0, -sp, 0, cp};
    const float Rz[9] = {cy, -sy, 0, sy, cy, 0, 0, 0, 1};
    float Ryx[9], Rj[9];
    mm3(Ry, Rx, Ryx);
    mm3(Rz, Ryx, Rj);     // R = Rz @ (Ry @ Rx)

    const float a0 = jaxis[3 * j + 0], a1 = jaxis[3 * j + 1], a2 = jaxis[3 * j + 2];
    const float n2 = a0 * a0 + a1 * a1 + a2 * a2;
    const float n  = sqrtf(n2);
    const float W[9] = {0, -a2, a1, a2, 0, -a0, -a1, a0, 0};
    float W2[9], RW[9], RW2[9];
    mm3(W, W, W2);
    mm3(Rj, W, RW);
    mm3(Rj, W2, RW2);
    const float inv_n = 1.0f / n, inv_n2 = 1.0f / n2;
    float* o = cst + j * CST_STRIDE;
#pragma unroll
    for (int i = 0; i < 9; ++i) {
      const float m2 = RW2[i] * inv_n2;
      o[i]      = Rj[i] + m2;    // P
      o[9 + i]  = RW[i] * inv_n; // Q
      o[18 + i] = -m2;           // Rm
    }
    o[27] = jxyz[3 * j + 0];
    o[28] = jxyz[3 * j + 1];
    o[29] = jxyz[3 * j + 2];
    o[30] = n;
    o[31] = 0.0f;
  } else if (j == NQ) {
    const int id = idx[0];
    cst[CST_LINK + 0] = lxyz[3 * id + 0];
    cst[CST_LINK + 1] = lxyz[3 * id + 1];
    cst[CST_LINK + 2] = lxyz[3 * id + 2];
  }
}

// ---------------- main: TDM-staged q, per-lane FMA chain -------------------

__global__ void __launch_bounds__(BLOCK)
fk_main_kernel(const float* __restrict__ q,
               const float* __restrict__ cst,
               float* __restrict__ out, int nb) {
  __shared__ float qs[BLOCK * NQ];

  const int b0 = blockIdx.x * BLOCK;

  // Wave 0 issues one Tensor-Data-Mover DMA: contiguous rem*NQ floats of q
  // for this block's bodies -> LDS.  (CDNA5 ISA)
  if (threadIdx.x < 32u) {
    int rem = nb - b0;
    if (rem > BLOCK) rem = BLOCK;
    const unsigned nelem = (unsigned)(rem * NQ);            // <= 1792, fits 16b tile dim
    const unsigned long long ga =
        (unsigned long long)(uintptr_t)(q + (size_t)b0 * NQ);
    const unsigned lds_addr = (unsigned)(uintptr_t)(&qs[0]); // low 32 bits = LDS offset

    v4u g0;
    g0.x = 1u;                                   // count=1 (valid), is_restore=0, gather=0
    g0.y = lds_addr;                             // lds_addr [63:32]
    g0.z = (unsigned)(ga & 0xFFFFFFFFull);       // global_addr lo   [95:64]
    g0.w = (unsigned)((ga >> 32) & 0x1FFFFFFull) // global_addr hi   [120:96]
           | (2u << 30);                         // type=2 ("image") [127:126]

    v8i g1;
    g1[0] = (int)(2u << 16);                         // data_size=4B; wg_mask=0
    g1[1] = (int)((nelem & 0xFFFFu) << 16);          // tensor_dim0 lo16
    g1[2] = (int)(((nelem >> 16) & 0xFFFFu) | (1u << 16)); // dim0 hi16 | tensor_dim1=1
    g1[3] = (int)((nelem & 0xFFFFu) << 16);          // tensor_dim1 hi=0 | tile_dim0
    g1[4] = 1;                                       // tile_dim1=1, tile_dim2=0
    g1[5] = (int)nelem;                              // tensor_dim0_stride lo32
    g1[6] = 0;                                       // stride hi | dim1_stride lo
    g1[7] = 0;                                       // dim1_stride hi

    v4i g2 = {0, 0, 0, 0};
    v4i g3 = {0, 0, 0, 0};
#if defined(__clang_major__) && (__clang_major__ >= 23)
    v8i g4 = {0, 0, 0, 0, 0, 0, 0, 0};
    __builtin_amdgcn_tensor_load_to_lds(g0, g1, g2, g3, g4, 0);
#else
    __builtin_amdgcn_tensor_load_to_lds(g0, g1, g2, g3, 0);
#endif
    __builtin_amdgcn_s_wait_tensorcnt(0);
  }
  __syncthreads();

  const int b = b0 + (int)threadIdx.x;
  if (b >= nb) return;

  float qv[NQ];
#pragma unroll
  for (int j = 0; j < NQ; ++j) qv[j] = qs[threadIdx.x * NQ + j];

  float Rw[9], tw[3];
  {
    const float* cj = cst;                 // joint 0: T_world = A_0, t_0
    float s, c;
    sincosf(cj[30] * qv[0], &s, &c);
#pragma unroll
    for (int i = 0; i < 9; ++i)
      Rw[i] = fmaf(c, cj[18 + i], fmaf(s, cj[9 + i], cj[i]));
    tw[0] = cj[27]; tw[1] = cj[28]; tw[2] = cj[29];
  }

#pragma unroll
  for (int j = 1; j < NQ; ++j) {
    const float* cj = cst + j * CST_STRIDE;
    float s, c;
    sincosf(cj[30] * qv[j], &s, &c);
    float A[9];
#pragma unroll
    for (int i = 0; i < 9; ++i)
      A[i] = fmaf(c, cj[18 + i], fmaf(s, cj[9 + i], cj[i]));

    // t_world = Rw @ t_j + t_world
    const float t0 = cj[27], t1 = cj[28], t2 = cj[29];
    tw[0] = fmaf(Rw[0], t0, fmaf(Rw[1], t1, fmaf(Rw[2], t2, tw[0])));
    tw[1] = fmaf(Rw[3], t0, fmaf(Rw[4], t1, fmaf(Rw[5], t2, tw[1])));
    tw[2] = fmaf(Rw[6], t0, fmaf(Rw[7], t1, fmaf(Rw[8], t2, tw[2])));

    // R_world = Rw @ A
    float Rn[9];
#pragma unroll
    for (int r = 0; r < 3; ++r)
#pragma unroll
      for (int cc = 0; cc < 3; ++cc)
        Rn[3 * r + cc] = fmaf(Rw[3 * r + 0], A[0 + cc],
                         fmaf(Rw[3 * r + 1], A[3 + cc],
                              Rw[3 * r + 2] * A[6 + cc]));
#pragma unroll
    for (int i = 0; i < 9; ++i) Rw[i] = Rn[i];
  }

  // T_world @ T_link (pure translation): t += Rw @ t_link
  {
    const float l0 = cst[CST_LINK + 0], l1 = cst[CST_LINK + 1], l2 = cst[CST_LINK + 2];
    tw[0] = fmaf(Rw[0], l0, fmaf(Rw[1], l1, fmaf(Rw[2], l2, tw[0])));
    tw[1] = fmaf(Rw[3], l0, fmaf(Rw[4], l1, fmaf(Rw[5], l2, tw[1])));
    tw[2] = fmaf(Rw[6], l0, fmaf(Rw[7], l1, fmaf(Rw[8], l2, tw[2])));
  }

  // Stream-out: write-once data, bypass temporal caching (th:NT) on the way
  // to HBM so the 32 MB result does not displace useful WGP$/L2 lines.
  v4f* o = reinterpret_cast<v4f*>(out + (size_t)b * 16);
  v4f r0 = {Rw[0], Rw[1], Rw[2], tw[0]};
  v4f r1 = {Rw[3], Rw[4], Rw[5], tw[1]};
  v4f r2 = {Rw[6], Rw[7], Rw[8], tw[2]};
  v4f r3 = {0.0f, 0.0f, 0.0f, 1.0f};
  __builtin_nontemporal_store(r0, o + 0);
  __builtin_nontemporal_store(r1, o + 1);
  __builtin_nontemporal_store(r2, o + 2);
  __builtin_nontemporal_store(r3, o + 3);
}

// ---------------- launch ---------------------------------------------------

extern "C" void kernel_launch(void* const* d_in, const int* in_sizes, int n_in,
                              void* d_out, int out_size, void* d_ws, size_t ws_size,
                              hipStream_t stream) {
  const float* q     = (const float*)d_in[0];
  const float* jxyz  = (const float*)d_in[1];
  const float* jrpy  = (const float*)d_in[2];
  const float* jaxis = (const float*)d_in[3];
  const float* lxyz  = (const float*)d_in[4];
  const int*   idx   = (const int*)d_in[5];
  float* out = (float*)d_out;
  float* cst = (float*)d_ws;

  const int nb = in_sizes[0] / NQ;

  fk_setup_kernel<<<1, 32, 0, stream>>>(jxyz, jrpy, jaxis, lxyz, idx, cst);

  const int blocks = (nb + BLOCK - 1) / BLOCK;
  fk_main_kernel<<<blocks, BLOCK, 0, stream>>>(q, cst, out, nb);
}